// LuongAttention_15272903704746
// MI455X (gfx1250) — compile-verified
//
#include <hip/hip_runtime.h>
#include <hip/hip_bf16.h>
#include <stdint.h>

// Luong attention, flash-style single HBM pass over enc_out (512 MB @ 23.3 TB/s ~= 22 us floor).
// Kernel 1: per (batch, S-segment) workgroup. Double-buffered async global->LDS staging of
//           16x1024 fp32 enc_out tiles (ASYNCcnt pipeline: issue tile i+1, wait <=16, compute
//           tile i), sims via chained V_WMMA_F32_16X16X4_F32 (two independent accumulator
//           chains for XDL ILP), online masked softmax, weighted context accumulation from
//           the same LDS tile (enc_out is read from HBM exactly once).
// Kernel 2: merges per-segment (m, l, acc) partials into context [B, D].

#define B_ 64
#define S_ 2048
#define D_ 1024
#define G_ 8                       // S-segments per batch
#define ROWS_PER_SEG (S_ / G_)     // 256
#define TILE_ 16                   // rows per LDS tile (matches WMMA M=16)
#define NTILES (ROWS_PER_SEG / TILE_)
#define TPAD 1028                  // padded LDS row stride (floats): 4112 B = 4096 + 16
#define NTHREADS 256

typedef float v2f __attribute__((ext_vector_type(2)));
typedef float v8f __attribute__((ext_vector_type(8)));

// Issue one 16-row tile: each thread moves 16 B of every row. The inst offset R*4096 is
// added to BOTH the global address and the LDS address; the extra +16 B/row of the padded
// LDS stride (4112 B) is folded into the vdst VGPR. One 64-bit global base per tile.
template <int R>
__device__ inline void issue_rows(uint32_t lds_base, const float* g_base) {
    asm volatile("global_load_async_to_lds_b128 %0, %1, off offset:%2"
                 :: "v"(lds_base + R * 16), "v"(g_base), "n"(R * 4096) : "memory");
    if constexpr (R + 1 < TILE_) issue_rows<R + 1>(lds_base, g_base);
}

__global__ __launch_bounds__(NTHREADS)
void attn_partial_kernel(const uint8_t* __restrict__ mask,   // [B,S] bool (1 byte)
                         const float*   __restrict__ enc,    // [B,S,D]
                         const float*   __restrict__ dec,    // [B,D]
                         float* __restrict__ ws_acc,          // [B*G, D]
                         float* __restrict__ ws_m,            // [B*G]
                         float* __restrict__ ws_l)            // [B*G]
{
    __shared__ float tile[2][TILE_][TPAD];  // 2 x 65792 B double-buffered enc_out tile
    __shared__ float sdec[D_];              // 4096 B dec_hid row
    __shared__ float part[8][16];           // per-wave partial sims
    __shared__ float srow[TILE_];           // masked sims for the tile
    __shared__ float wrow[TILE_];           // softmax weights for the tile

    const int wg   = blockIdx.x;
    const int b    = wg / G_;
    const int g    = wg % G_;
    const int t    = threadIdx.x;
    const int lane = t & 31;
    const int wave = t >> 5;
    const int s_base = g * ROWS_PER_SEG;
    const int col  = t * 4;               // this thread owns context columns col..col+3

    const float*   encB  = enc  + ((size_t)b * S_ + s_base) * D_;
    const float*   decB  = dec  + (size_t)b * D_;
    const uint8_t* maskB = mask + (size_t)b * S_ + s_base;

    // Preload dec_hid[b] into LDS (visible after the first tile barrier).
    *(float4*)&sdec[col] = *(const float4*)(decB + col);

    float m_run = -INFINITY;
    float l_run = 0.f;
    float acc0 = 0.f, acc1 = 0.f, acc2 = 0.f, acc3 = 0.f;

    // Prologue: stage tile 0 into buffer 0.
    issue_rows<0>((uint32_t)(size_t)(&tile[0][0][col]), encB + col);

    for (int ti = 0; ti < NTILES; ++ti) {
        const int cur = ti & 1;
        const int r0  = ti * TILE_;

        // Prefetch tile ti+1 into the other buffer, then wait for tile ti only.
        if (ti + 1 < NTILES) {
            issue_rows<0>((uint32_t)(size_t)(&tile[cur ^ 1][0][col]),
                          encB + (size_t)(ti + 1) * TILE_ * D_ + col);
            // Async loads complete in order: <=16 outstanding => tile ti has landed.
            asm volatile("s_wait_asynccnt 0x10" ::: "memory");
        } else {
            asm volatile("s_wait_asynccnt 0x0" ::: "memory");
        }
        __syncthreads();

        const float (*tl)[TPAD] = tile[cur];

        // ---- Phase 1: sims for the 16 rows via chained V_WMMA_F32_16X16X4_F32.
        //      Wave w reduces K in [w*128, w*128+128), split over two independent
        //      accumulator chains. A (16x4 f32): lanes 0-15 hold K={0,1}, lanes 16-31
        //      K={2,3}; B mirrors it with dec_hid broadcast into all 16 N columns,
        //      so D[m,n] == sims[m] for every n.
        {
            v8f c0 = {0.f,0.f,0.f,0.f,0.f,0.f,0.f,0.f};
            v8f c1 = {0.f,0.f,0.f,0.f,0.f,0.f,0.f,0.f};
            const int row = lane & 15;
            const int kg  = (lane >> 4) << 1;   // 0 or 2
            const int kw  = wave * 128;
            #pragma unroll
            for (int kb = 0; kb < 128; kb += 8) {
                const int k0 = kw + kb + kg;
                v2f a0 = { tl[row][k0], tl[row][k0 + 1] };
                v2f b0 = { sdec[k0],    sdec[k0 + 1] };
                c0 = __builtin_amdgcn_wmma_f32_16x16x4_f32(
                         false, a0, false, b0, (short)0, c0, false, false);
                const int k1 = k0 + 4;
                v2f a1 = { tl[row][k1], tl[row][k1 + 1] };
                v2f b1 = { sdec[k1],    sdec[k1 + 1] };
                c1 = __builtin_amdgcn_wmma_f32_16x16x4_f32(
                         false, a1, false, b1, (short)0, c1, false, false);
            }
            // Column N=0: lane 0 holds M=0..7, lane 16 holds M=8..15.
            if (lane == 0) {
                #pragma unroll
                for (int i = 0; i < 8; ++i) part[wave][i] = c0[i] + c1[i];
            } else if (lane == 16) {
                #pragma unroll
                for (int i = 0; i < 8; ++i) part[wave][8 + i] = c0[i] + c1[i];
            }
        }
        __syncthreads();

        // ---- Phase 2: reduce wave partials, apply padding mask.
        if (t < TILE_) {
            float s = 0.f;
            #pragma unroll
            for (int w = 0; w < 8; ++w) s += part[w][t];
            srow[t] = (maskB[r0 + t] != 0) ? -INFINITY : s;
        }
        __syncthreads();

        // Online softmax update (every thread computes the same scalars redundantly).
        float m_tile = -INFINITY;
        #pragma unroll
        for (int r = 0; r < TILE_; ++r) m_tile = fmaxf(m_tile, srow[r]);
        float m_new = fmaxf(m_run, m_tile);
        float scale = (m_run == -INFINITY) ? 0.f : __expf(m_run - m_new);
        if (t < TILE_) {
            wrow[t] = (srow[t] == -INFINITY) ? 0.f : __expf(srow[t] - m_new);
        }
        __syncthreads();

        float wsum = 0.f;
        #pragma unroll
        for (int r = 0; r < TILE_; ++r) wsum += wrow[r];
        l_run = l_run * scale + wsum;
        m_run = m_new;

        // ---- Phase 3: weighted accumulation from the SAME LDS tile (no 2nd HBM pass).
        acc0 *= scale; acc1 *= scale; acc2 *= scale; acc3 *= scale;
        #pragma unroll
        for (int r = 0; r < TILE_; ++r) {
            float  w = wrow[r];
            float4 e = *(const float4*)&tl[r][col];
            acc0 = fmaf(w, e.x, acc0);
            acc1 = fmaf(w, e.y, acc1);
            acc2 = fmaf(w, e.z, acc2);
            acc3 = fmaf(w, e.w, acc3);
        }
        __syncthreads();   // tile consumed; its buffer may be refilled next iteration
    }

    float* accOut = ws_acc + (size_t)wg * D_ + col;
    accOut[0] = acc0; accOut[1] = acc1; accOut[2] = acc2; accOut[3] = acc3;
    if (t == 0) { ws_m[wg] = m_run; ws_l[wg] = l_run; }
}

__global__ __launch_bounds__(NTHREADS)
void attn_combine_kernel(const float* __restrict__ ws_acc,
                         const float* __restrict__ ws_m,
                         const float* __restrict__ ws_l,
                         float* __restrict__ out)            // [B, D]
{
    const int b = blockIdx.x;
    const int t = threadIdx.x;
    const int col = t * 4;

    float m[G_], l[G_], w[G_];
    float M = -INFINITY;
    #pragma unroll
    for (int g = 0; g < G_; ++g) {
        m[g] = ws_m[b * G_ + g];
        l[g] = ws_l[b * G_ + g];
        M = fmaxf(M, m[g]);
    }
    float L = 0.f;
    #pragma unroll
    for (int g = 0; g < G_; ++g) {
        w[g] = (m[g] == -INFINITY) ? 0.f : __expf(m[g] - M);
        L += w[g] * l[g];
    }
    const float inv = (L > 0.f) ? (1.f / L) : 0.f;

    float sx = 0.f, sy = 0.f, sz = 0.f, sw = 0.f;
    #pragma unroll
    for (int g = 0; g < G_; ++g) {
        float4 a = *(const float4*)(ws_acc + ((size_t)(b * G_ + g)) * D_ + col);
        sx = fmaf(w[g], a.x, sx);
        sy = fmaf(w[g], a.y, sy);
        sz = fmaf(w[g], a.z, sz);
        sw = fmaf(w[g], a.w, sw);
    }
    float4 o = { sx * inv, sy * inv, sz * inv, sw * inv };
    *(float4*)(out + (size_t)b * D_ + col) = o;
}

extern "C" void kernel_launch(void* const* d_in, const int* in_sizes, int n_in,
                              void* d_out, int out_size, void* d_ws, size_t ws_size,
                              hipStream_t stream) {
    (void)in_sizes; (void)n_in; (void)out_size; (void)ws_size;
    // setup_inputs order: enc_mask [B,S] bool(1B), enc_out [B,S,D] f32, dec_hid [B,D] f32
    const uint8_t* mask = (const uint8_t*)d_in[0];
    const float*   enc  = (const float*)d_in[1];
    const float*   dec  = (const float*)d_in[2];
    float* out = (float*)d_out;

    // Workspace carve-out: acc[B*G*D] | m[B*G] | l[B*G]  (~2.1 MB)
    float* ws_acc = (float*)d_ws;
    float* ws_m   = ws_acc + (size_t)B_ * G_ * D_;
    float* ws_l   = ws_m + (size_t)B_ * G_;

    attn_partial_kernel<<<dim3(B_ * G_), dim3(NTHREADS), 0, stream>>>(
        mask, enc, dec, ws_acc, ws_m, ws_l);
    attn_combine_kernel<<<dim3(B_), dim3(NTHREADS), 0, stream>>>(
        ws_acc, ws_m, ws_l, out);
}